// FourierConv_43963285241881
// MI455X (gfx1250) — compile-verified
//
#include <hip/hip_runtime.h>

// FourierConv on MI455X (gfx1250):
//   y[b,h,w,c] = Re( IFFT2( FFT2(pad(x[b,:,:,c])) * Ksum[:,:,c] ) )[h+63,w+63] + bias[c]
// with Ksum = sum over COUT of (kernel_real + i*kernel_imag).
// DFT-255 expressed as dense matmuls padded to 256, on V_WMMA_F32_16X16X4_F32.
// Memory floor: ~133 MB kernel reads in reduce_k (~5.7 us @ 23.3 TB/s).

typedef __attribute__((ext_vector_type(2))) float v2f;
typedef __attribute__((ext_vector_type(8))) float v8f;

#define PI_F 3.14159265358979323846f

// --- explicit global-address-space accessors: force global_load/global_store
// (round 2 showed pointer arrays demoting A loads to flat_load_b64 + DScnt) ---
typedef __attribute__((address_space(1))) float       gfloat_t;
typedef const __attribute__((address_space(1))) float gcfloat_t;
typedef const __attribute__((address_space(1))) v2f   gcv2f_t;

static __device__ __forceinline__ v2f gload2(const float* p) {
    return *(gcv2f_t*)(unsigned long long)p;
}
static __device__ __forceinline__ float gload1(const float* p) {
    return *(gcfloat_t*)(unsigned long long)p;
}
static __device__ __forceinline__ void gstore1(float* p, float v) {
    *(gfloat_t*)(unsigned long long)p = v;
}

static __device__ __forceinline__ v8f wmma4(v2f a, v2f b, v8f c) {
    // D = A(16x4) * B(4x16) + C(16x16), all f32
    return __builtin_amdgcn_wmma_f32_16x16x4_f32(
        false, a, false, b, (short)0, c, false, false);
}

// ---------------------------------------------------------------------------
// Twiddle builders. F[u,h] = exp(-2*pi*i*u*h/255), u in [0,256) (row 255 = 0),
// h in [0,128). G[h,u] = (1/255^2)*exp(+2*pi*i*u*(h+63)/255), col u=255 = 0.
// Phase reduced mod 255 in integers to keep sin/cos arguments small.
// ---------------------------------------------------------------------------
__global__ __launch_bounds__(256) void build_F(float* __restrict__ Fr,
                                               float* __restrict__ Fi) {
    int idx = blockIdx.x * 256 + threadIdx.x;          // 256*128
    int u = idx >> 7, h = idx & 127;
    float re = 0.f, im = 0.f;
    if (u < 255) {
        int p = (u * h) % 255;
        float th = -2.0f * PI_F * (float)p * (1.0f / 255.0f);
        re = cosf(th); im = sinf(th);
    }
    Fr[idx] = re; Fi[idx] = im;
}

__global__ __launch_bounds__(256) void build_G(float* __restrict__ Gr,
                                               float* __restrict__ Gi) {
    int idx = blockIdx.x * 256 + threadIdx.x;          // 128*256
    int h = idx >> 8, u = idx & 255;
    float re = 0.f, im = 0.f;
    if (u < 255) {
        int q = (u * (h + 63)) % 255;
        float th = 2.0f * PI_F * (float)q * (1.0f / 255.0f);
        float s = 1.0f / (255.0f * 255.0f);
        re = s * cosf(th); im = s * sinf(th);
    }
    Gr[idx] = re; Gi[idx] = im;
}

// ---------------------------------------------------------------------------
// Ksum[c][u][v] = sum_o kernel[u,v,c,o], padded to 256x256 (zeros at 255).
// Each thread owns one (u,v): reads 2x1KB contiguous, writes 16 planes.
// This is the memory-bound stage: ~133 MB of kernel reads.
// ---------------------------------------------------------------------------
__global__ __launch_bounds__(256) void reduce_k(const float* __restrict__ kr,
                                                const float* __restrict__ ki,
                                                float* __restrict__ Ksr,
                                                float* __restrict__ Ksi) {
    int idx = blockIdx.x * 256 + threadIdx.x;          // 256*256
    int u = idx >> 8, v = idx & 255;
    float sr[16], si[16];
#pragma unroll
    for (int c = 0; c < 16; ++c) { sr[c] = 0.f; si[c] = 0.f; }
    if (u < 255 && v < 255) {
        const float4* pr = (const float4*)(kr + (size_t)(u * 255 + v) * 256);
        const float4* pi = (const float4*)(ki + (size_t)(u * 255 + v) * 256);
#pragma unroll
        for (int c = 0; c < 16; ++c) {
            float aR = 0.f, aI = 0.f;
#pragma unroll
            for (int q = 0; q < 4; ++q) {
                float4 r4 = pr[c * 4 + q], i4 = pi[c * 4 + q];
                aR += r4.x + r4.y + r4.z + r4.w;
                aI += i4.x + i4.y + i4.z + i4.w;
            }
            sr[c] = aR; si[c] = aI;
        }
    }
#pragma unroll
    for (int c = 0; c < 16; ++c) {
        Ksr[((size_t)c << 16) + idx] = sr[c];
        Ksi[((size_t)c << 16) + idx] = si[c];
    }
}

// ---------------------------------------------------------------------------
// Gather x (B,H,W,CIN) -> 64 contiguous 128x128 slices, slice s = b*16+c.
// ---------------------------------------------------------------------------
__global__ __launch_bounds__(256) void gather_x(const float* __restrict__ x,
                                                float* __restrict__ Xs) {
    int idx = blockIdx.x * 256 + threadIdx.x;          // 4*128*128*16
    int c = idx & 15;
    int w = (idx >> 4) & 127;
    int h = (idx >> 11) & 127;
    int b = idx >> 18;
    Xs[(((size_t)(b * 16 + c)) << 14) + (h << 7) + w] = x[idx];
}

// ---------------------------------------------------------------------------
// Batched complex matmul D = A * B over 64 slices (blockIdx.z).
// Wave = 64x16 tile (4 row-tiles reuse each B fragment -> 16 WMMA per B load);
// block = (32,4): 4 waves over N -> block tile 64x64.
// LDA/LDB/LDD are compile-time so row offsets fold into load/store immediates
// and the k-loop bumps exactly one pointer per plane.
// BREAL : B imag plane absent
// BTRANS: B accessed as B[k][n] = Bmem[n*LDB + k] (contiguous k -> v2f load)
// EPI   : 0 = store both planes, 1 = multiply by Ksum[c] then store,
//         2 = real part + bias -> d_out (B,H,W,CIN layout; imag never computed)
// ---------------------------------------------------------------------------
template <bool BREAL, bool BTRANS, int EPI, int LDA, int LDB, int LDD>
__global__ __launch_bounds__(128) void cmatmul(
    const float* __restrict__ Ar, const float* __restrict__ Ai, int sA,
    const float* __restrict__ Br, const float* __restrict__ Bi, int sB,
    float* __restrict__ Dr, float* __restrict__ Di, int sD, int K,
    const float* __restrict__ ksr, const float* __restrict__ ksi,
    const float* __restrict__ bias, float* __restrict__ out) {
    const int s  = blockIdx.z;
    const int n0 = blockIdx.x * 64 + threadIdx.y * 16;
    const int m0 = blockIdx.y * 64;
    const int lane = threadIdx.x;
    const int lm = lane & 15;
    const int lk = (lane >> 4) << 1;                   // 0 or 2

    // single base pointer per plane; row-tile offsets are immediates
    const float* pa = Ar + (size_t)s * sA + (size_t)(m0 + lm) * LDA + lk;
    const float* pi = Ai + (size_t)s * sA + (size_t)(m0 + lm) * LDA + lk;
    const float* pbr;
    const float* pbi = nullptr;
    if (BTRANS) {
        pbr = Br + (size_t)s * sB + (size_t)(n0 + lm) * LDB + lk;
        if (!BREAL) pbi = Bi + (size_t)s * sB + (size_t)(n0 + lm) * LDB + lk;
    } else {
        pbr = Br + (size_t)s * sB + (size_t)lk * LDB + (n0 + lm);
        if (!BREAL) pbi = Bi + (size_t)s * sB + (size_t)lk * LDB + (n0 + lm);
    }

    v8f accR[4], accI[4];
#pragma unroll
    for (int rt = 0; rt < 4; ++rt) {
        accR[rt] = (v8f){0.f, 0.f, 0.f, 0.f, 0.f, 0.f, 0.f, 0.f};
        accI[rt] = (v8f){0.f, 0.f, 0.f, 0.f, 0.f, 0.f, 0.f, 0.f};
    }

    for (int k0 = 0; k0 < K; k0 += 4) {
        v2f bR, bI = {0.f, 0.f};
        if (BTRANS) {
            bR = gload2(pbr);
            if (!BREAL) bI = gload2(pbi);
            __builtin_prefetch(pbr + 32, 0, 3);        // global_prefetch_b8
            pbr += 4;
            if (!BREAL) pbi += 4;
        } else {
            bR.x = gload1(pbr);
            bR.y = gload1(pbr + LDB);
            if (!BREAL) { bI.x = gload1(pbi); bI.y = gload1(pbi + LDB); }
            __builtin_prefetch(pbr + 8 * LDB, 0, 3);
            pbr += 4 * LDB;
            if (!BREAL) pbi += 4 * LDB;
        }
#pragma unroll
        for (int rt = 0; rt < 4; ++rt) {
            v2f aR = gload2(pa + rt * 16 * LDA);       // offset:<imm>
            v2f aI = gload2(pi + rt * 16 * LDA);
            accR[rt] = wmma4(aR, bR, accR[rt]);        // Re += Ar*Br
            if (!BREAL) {
                v2f aIn = -aI;                         // f32 WMMA has no A-neg
                accR[rt] = wmma4(aIn, bI, accR[rt]);   // Re -= Ai*Bi
            }
            if (EPI != 2) {
                accI[rt] = wmma4(aI, bR, accI[rt]);    // Im += Ai*Br
                if (!BREAL) accI[rt] = wmma4(aR, bI, accI[rt]); // Im += Ar*Bi
            }
        }
        pa += 4;
        pi += 4;
    }

    const int nn = n0 + lm;
    const int mhi = (lane >> 4) << 3;                   // +8 for lanes 16..31
    const int c = s & 15;
    if (EPI == 0 || EPI == 1) {
        float* pdr = Dr + (size_t)s * sD + (size_t)(m0 + mhi) * LDD + nn;
        float* pdi = Di + (size_t)s * sD + (size_t)(m0 + mhi) * LDD + nn;
        const float* pkr = ksr + ((size_t)c << 16) + ((size_t)(m0 + mhi) << 8) + nn;
        const float* pki = ksi + ((size_t)c << 16) + ((size_t)(m0 + mhi) << 8) + nn;
#pragma unroll
        for (int rt = 0; rt < 4; ++rt) {
#pragma unroll
            for (int r = 0; r < 8; ++r) {
                const int off = (rt * 16 + r) * LDD;
                float vr = accR[rt][r], vi = accI[rt][r];
                if (EPI == 0) {
                    gstore1(pdr + off, vr);
                    gstore1(pdi + off, vi);
                } else {
                    const int koff = (rt * 16 + r) * 256;
                    const float kr = gload1(pkr + koff), kiv = gload1(pki + koff);
                    gstore1(pdr + off, vr * kr - vi * kiv);
                    gstore1(pdi + off, vr * kiv + vi * kr);
                }
            }
        }
    } else {
        const int b = s >> 4;
        float* po = out + (((size_t)b * 128 + (m0 + mhi)) * 128 + nn) * 16 + c;
        const float bv = gload1(bias + c);
#pragma unroll
        for (int rt = 0; rt < 4; ++rt) {
#pragma unroll
            for (int r = 0; r < 8; ++r) {
                gstore1(po + (rt * 16 + r) * 128 * 16, accR[rt][r] + bv);
            }
        }
    }
}

// ---------------------------------------------------------------------------
extern "C" void kernel_launch(void* const* d_in, const int* in_sizes, int n_in,
                              void* d_out, int out_size, void* d_ws, size_t ws_size,
                              hipStream_t stream) {
    const float* x    = (const float*)d_in[0];
    const float* kre  = (const float*)d_in[1];
    const float* kim  = (const float*)d_in[2];
    const float* bias = (const float*)d_in[3];
    float* out = (float*)d_out;

    float* ws  = (float*)d_ws;
    float* Fr  = ws;                    // 256*128
    float* Fi  = Fr + 32768;
    float* Gr  = Fi + 32768;            // 128*256
    float* Gi  = Gr + 32768;
    float* Ksr = Gi + 32768;            // 16*256*256
    float* Ksi = Ksr + 16 * 65536;
    float* Xs  = Ksi + 16 * 65536;      // 64*128*128
    float* Ar  = Xs + 64 * 16384;       // 64*256*128
    float* Ai  = Ar + 64 * 32768;
    float* Zr  = Ai + 64 * 32768;       // 64*256*256
    float* Zi  = Zr + 64 * 65536;
    float* Tr  = Zi + 64 * 65536;       // 64*128*256
    float* Ti  = Tr + 64 * 32768;

    build_F<<<128, 256, 0, stream>>>(Fr, Fi);
    build_G<<<128, 256, 0, stream>>>(Gr, Gi);
    reduce_k<<<256, 256, 0, stream>>>(kre, kim, Ksr, Ksi);
    gather_x<<<4096, 256, 0, stream>>>(x, Xs);

    dim3 blk(32, 4, 1);
    // A = F (256x128 cplx) * x_s (128x128 real)          M=256 N=128 K=128
    cmatmul<true, false, 0, 128, 128, 128><<<dim3(2, 4, 64), blk, 0, stream>>>(
        Fr, Fi, 0, Xs, nullptr, 16384, Ar, Ai, 32768, 128,
        nullptr, nullptr, nullptr, nullptr);
    // Z = (A * F^T) .* Ksum[c]                           M=256 N=256 K=128
    cmatmul<false, true, 1, 128, 128, 256><<<dim3(4, 4, 64), blk, 0, stream>>>(
        Ar, Ai, 32768, Fr, Fi, 0, Zr, Zi, 65536, 128,
        Ksr, Ksi, nullptr, nullptr);
    // T = G (128x256 cplx) * Z (256x256 cplx)            M=128 N=256 K=256
    cmatmul<false, false, 0, 256, 256, 256><<<dim3(4, 2, 64), blk, 0, stream>>>(
        Gr, Gi, 0, Zr, Zi, 65536, Tr, Ti, 32768, 256,
        nullptr, nullptr, nullptr, nullptr);
    // y = Re(T * G^T) + bias -> d_out (B,H,W,CIN)        M=128 N=128 K=256
    cmatmul<false, true, 2, 256, 256, 1><<<dim3(2, 2, 64), blk, 0, stream>>>(
        Tr, Ti, 32768, Gr, Gi, 0, nullptr, nullptr, 0, 256,
        nullptr, nullptr, bias, out);
}